// LocalAttention_86225763435394
// MI455X (gfx1250) — compile-verified
//
#include <hip/hip_runtime.h>
#include <hip/hip_bf16.h>
#include <math.h>

typedef __attribute__((ext_vector_type(2))) float v2f;
typedef __attribute__((ext_vector_type(8))) float v8f;

#define BDIM 256
#define LDIM 500
#define EDIM 300
#define NROWS (BDIM * LDIM)   // 128000
#define NCOL 18               // compact D columns: 0-2 aw3, 3-7 aw5, 8-14 aw7, 15 cw3, 16 cw5, 17 cw7

// -----------------------------------------------------------------------------
// Kernel 1: D = X (128000 x 300) * W (300 x 32-padded), tall-skinny WMMA GEMM.
// Each wave owns a 16-row tile with two 16x16 f32 accumulators:
//   tile0 cols 0-14 = conv weight rows (aw3|aw5|aw7), col 15 pad
//   tile1 cols 0-2  = cw3/cw5/cw7, rest pad
// K loop: 75 x V_WMMA_F32_16X16X4_F32 per tile (full fp32, K=4/step).
//
// B operand LDS layout (K-pair interleaved + bank swizzle):
//   wlds[kp*64 + ((col*2 + j) ^ ((kp&1)<<5))] = W[2*kp + j][col]
// so a B fragment {W[ko][col], W[ko+1][col]} is ONE ds_load_b64, and the
// kp-parity XOR puts lanes 0-15 (even kp) and 16-31 (odd kp) in disjoint
// bank halves -> conflict-free.  kp parity is loop-invariant per lane, so the
// swizzle folds into a base pointer and the loop uses immediate DS offsets.
// -----------------------------------------------------------------------------
__global__ __launch_bounds__(256) void la_gemm(
    const float* __restrict__ x,
    const float* __restrict__ aw3, const float* __restrict__ aw5,
    const float* __restrict__ aw7,
    const float* __restrict__ cw3, const float* __restrict__ cw5,
    const float* __restrict__ cw7,
    float* __restrict__ D)
{
    __shared__ float wlds[(EDIM / 2) * 64];   // 150 kp-rows x 64 floats = 38.4 KB

    const int tid = threadIdx.x;

    // Pack W into LDS in the paired+swizzled layout.
    for (int i = tid; i < (EDIM / 2) * 64; i += 256) {
        const int kp   = i >> 6;
        const int f    = i & 63;
        const int colj = f ^ ((kp & 1) << 5);
        const int col  = colj >> 1;
        const int k    = kp * 2 + (colj & 1);
        float v = 0.0f;
        if      (col < 3)   v = aw3[col * EDIM + k];
        else if (col < 8)   v = aw5[(col - 3) * EDIM + k];
        else if (col < 15)  v = aw7[(col - 8) * EDIM + k];
        else if (col == 16) v = cw3[k];
        else if (col == 17) v = cw5[k];
        else if (col == 18) v = cw7[k];
        wlds[i] = v;
    }
    __syncthreads();

    const int lane = tid & 31;
    const int wave = tid >> 5;
    const int half = lane >> 4;   // 0: lanes 0-15 (K+0/1, M 0-7), 1: lanes 16-31 (K+2/3, M 8-15)
    const int n    = lane & 15;   // A: row-in-tile; B/C: column-in-tile

    const long rowBase = ((long)blockIdx.x * 8 + wave) * 16;

    // A fragment source: x[rowBase + n][kb + 2*half + {0,1}], one b64 per step.
    const float* ap = x + (rowBase + n) * EDIM + 2 * half;

    // Loop-invariant B fragment bases (kp = kb/2 + half  ->  float idx kb*32 + base).
    const int sw = half << 5;
    const float* bp0 = wlds + (half << 6) + (((n     ) * 2) ^ sw);
    const float* bp1 = wlds + (half << 6) + (((n + 16) * 2) ^ sw);

    v8f c0 = {};  // conv columns
    v8f c1 = {};  // cw columns

    for (int kb = 0; kb < EDIM; kb += 4) {
        v2f a  = *(const v2f*)(ap  + kb);
        v2f b0 = *(const v2f*)(bp0 + kb * 32);
        v2f b1 = *(const v2f*)(bp1 + kb * 32);
        c0 = __builtin_amdgcn_wmma_f32_16x16x4_f32(false, a, false, b0,
                                                   (short)0, c0, false, false);
        c1 = __builtin_amdgcn_wmma_f32_16x16x4_f32(false, a, false, b1,
                                                   (short)0, c1, false, false);
    }

    // Scatter the 16x18 useful part of the 16x32 tile to the compact D matrix.
    float* drow = D + rowBase * NCOL;
#pragma unroll
    for (int v = 0; v < 8; ++v) {
        const int m = v + 8 * half;          // C layout: lanes>=16 hold M=8..15
        if (n < 15) drow[m * NCOL + n]      = c0[v];   // conv columns 0..14
        if (n < 3)  drow[m * NCOL + 15 + n] = c1[v];   // cw columns 15..17
    }
}

// -----------------------------------------------------------------------------
// Kernel 2: stencil + activation epilogue.  One thread per (b,l) position.
//   s_k[r] = sum_dk D[r + dk - p, col(k,dk)]   (zero-padded at L boundaries)
//   out_k  = tanh( sigmoid(s_k + ab_k) * D[r, cw_col_k] + cb_k )
// -----------------------------------------------------------------------------
__global__ __launch_bounds__(256) void la_epilogue(
    const float* __restrict__ D,
    const float* __restrict__ ab3, const float* __restrict__ cb3,
    const float* __restrict__ ab5, const float* __restrict__ cb5,
    const float* __restrict__ ab7, const float* __restrict__ cb7,
    float* __restrict__ out)
{
    const int r = blockIdx.x * blockDim.x + threadIdx.x;
    if (r >= NROWS) return;
    const int l = r % LDIM;
    const float* dr = D + (long)r * NCOL;

    float s3 = 0.0f, s5 = 0.0f, s7 = 0.0f;
#pragma unroll
    for (int dk = 0; dk < 3; ++dk) {
        const int lo = l + dk - 1;
        if (lo >= 0 && lo < LDIM) s3 += dr[(long)(dk - 1) * NCOL + dk];
    }
#pragma unroll
    for (int dk = 0; dk < 5; ++dk) {
        const int lo = l + dk - 2;
        if (lo >= 0 && lo < LDIM) s5 += dr[(long)(dk - 2) * NCOL + 3 + dk];
    }
#pragma unroll
    for (int dk = 0; dk < 7; ++dk) {
        const int lo = l + dk - 3;
        if (lo >= 0 && lo < LDIM) s7 += dr[(long)(dk - 3) * NCOL + 8 + dk];
    }

    const float xc3 = dr[15], xc5 = dr[16], xc7 = dr[17];

    const float sc3 = 1.0f / (1.0f + expf(-(s3 + ab3[0])));
    const float sc5 = 1.0f / (1.0f + expf(-(s5 + ab5[0])));
    const float sc7 = 1.0f / (1.0f + expf(-(s7 + ab7[0])));

    out[r]             = tanhf(sc3 * xc3 + cb3[0]);
    out[NROWS + r]     = tanhf(sc5 * xc5 + cb5[0]);
    out[2 * NROWS + r] = tanhf(sc7 * xc7 + cb7[0]);
}

extern "C" void kernel_launch(void* const* d_in, const int* in_sizes, int n_in,
                              void* d_out, int out_size, void* d_ws, size_t ws_size,
                              hipStream_t stream) {
    const float* x   = (const float*)d_in[0];
    const float* aw3 = (const float*)d_in[1];
    const float* ab3 = (const float*)d_in[2];
    const float* cw3 = (const float*)d_in[3];
    const float* cb3 = (const float*)d_in[4];
    const float* aw5 = (const float*)d_in[5];
    const float* ab5 = (const float*)d_in[6];
    const float* cw5 = (const float*)d_in[7];
    const float* cb5 = (const float*)d_in[8];
    const float* aw7 = (const float*)d_in[9];
    const float* ab7 = (const float*)d_in[10];
    const float* cw7 = (const float*)d_in[11];
    const float* cb7 = (const float*)d_in[12];

    float* D   = (float*)d_ws;    // needs NROWS*18*4 = 9.2 MB of scratch
    float* out = (float*)d_out;

    // 1000 blocks x 8 waves x 16 rows = 128000 rows exactly.
    la_gemm<<<NROWS / 128, 256, 0, stream>>>(x, aw3, aw5, aw7, cw3, cw5, cw7, D);
    la_epilogue<<<(NROWS + 255) / 256, 256, 0, stream>>>(D, ab3, cb3, ab5, cb5,
                                                         ab7, cb7, out);
}